// MAE3DEncoder_67104569033404
// MI455X (gfx1250) — compile-verified
//
#include <hip/hip_runtime.h>
#include <hip/hip_bf16.h>
#include <math.h>

typedef __attribute__((ext_vector_type(2))) float v2f;
typedef __attribute__((ext_vector_type(8))) float v8f;

#define BN_INV_F   0.99999500003749968f   /* 1/sqrt(1+1e-5) */
#define LRELU(x)   ((x) > 0.f ? (x) : 0.2f*(x))
#define NEG_INF    (-3.0e38f)

__device__ __forceinline__ v8f wmma_f32(v2f a, v2f b, v8f c) {
  // D = A(16x4,f32) * B(4x16,f32) + C(16x16,f32), wave32
  return __builtin_amdgcn_wmma_f32_16x16x4_f32(false, a, false, b, (short)0, c, false, false);
}

// ---------------------------------------------------------------------------
// kNN top-20 (largest pd = -||xi-xj||^2), one wave per query point.
// F is (B*2048, ldF) row-major; features at columns [colF, colF+CIN).
// ---------------------------------------------------------------------------
template<int CIN>
__global__ __launch_bounds__(32)
void knn_kernel(const float* __restrict__ F, int ldF, int colF, int* __restrict__ idx_out)
{
  const int NPTS = 2048, KNN = 20;
  int wg   = blockIdx.x;        // b*NPTS + n
  int b    = wg >> 11;
  int lane = threadIdx.x;

  __shared__ float s_q[CIN];
  __shared__ float s_lv[KNN*32];
  __shared__ int   s_li[KNN*32];

  const float* qrow = F + (size_t)wg*ldF + colF;
  for (int c = lane; c < CIN; c += 32) s_q[c] = qrow[c];
  __syncthreads();

  float qq = 0.f;
  #pragma unroll 8
  for (int c = 0; c < CIN; ++c) qq += s_q[c]*s_q[c];

  const float* baseF = F + (size_t)(b*NPTS)*ldF + colF;
  int cnt = 0;
  for (int m = lane; m < NPTS; m += 32) {
    const float* fr = baseF + (size_t)m*ldF;
    float dot = 0.f, mm = 0.f;
    #pragma unroll 8
    for (int c = 0; c < CIN; ++c) { float xv = fr[c]; dot += s_q[c]*xv; mm += xv*xv; }
    float pd = 2.f*dot - qq - mm;
    if (cnt == KNN && pd <= s_lv[(KNN-1)*32 + lane]) continue;
    int j = (cnt < KNN) ? cnt : (KNN-1);
    while (j > 0 && s_lv[(j-1)*32 + lane] < pd) {
      s_lv[j*32+lane] = s_lv[(j-1)*32+lane];
      s_li[j*32+lane] = s_li[(j-1)*32+lane];
      --j;
    }
    s_lv[j*32+lane] = pd;
    s_li[j*32+lane] = m;
    if (cnt < KNN) ++cnt;
  }

  // wave-wide merge of 32 sorted lists -> global top-20 (ties: smaller index)
  int ptr = 0;
  int* out = idx_out + (size_t)wg*KNN;
  for (int r = 0; r < KNN; ++r) {
    float v  = (ptr < cnt) ? s_lv[ptr*32+lane] : NEG_INF;
    int   id = (ptr < cnt) ? s_li[ptr*32+lane] : 0x7fffffff;
    float bv = v; int bi = id;
    #pragma unroll
    for (int off = 16; off > 0; off >>= 1) {
      float ov = __shfl_xor(bv, off);
      int   oi = __shfl_xor(bi, off);
      if (ov > bv || (ov == bv && oi < bi)) { bv = ov; bi = oi; }
    }
    if (v == bv && id == bi) ++ptr;
    if (lane == 0) out[r] = bi;
  }
}

// ---------------------------------------------------------------------------
// Edge-conv: one block per point, COUT/16 waves; edge matrix (32 x C2P,
// zero-padded) built once in LDS, then each wave does a 16-wide O slice with
// fp32 WMMA. BN+LeakyReLU+max-over-k fused, branchless, in accumulators.
// ---------------------------------------------------------------------------
template<int CIN, int COUT>
__global__ __launch_bounds__((COUT/16)*32)
void edgeconv_kernel(const float* __restrict__ F, int ldF, int colF,
                     const int* __restrict__ idx,
                     const float* __restrict__ W,    // (COUT, 2*CIN) row-major
                     const float* __restrict__ bn,   // gamma, beta
                     float* __restrict__ Out, int ldOut, int colOut)
{
  constexpr int NPTS = 2048, KNN = 20;
  constexpr int C2  = 2*CIN;
  constexpr int C2P = (C2 + 3) & ~3;           // pad K-dim to multiple of 4
  constexpr int NW  = COUT/16;
  constexpr bool PADW = (C2P != C2);

  int wg   = blockIdx.x;                       // b*NPTS + n
  int b    = wg >> 11;
  int tid  = threadIdx.x;
  int lane = tid & 31;
  int wv   = tid >> 5;

  __shared__ float s_edge[32*C2P];             // rows 20..31 & cols>=C2 zero
  __shared__ float s_cen[CIN];
  __shared__ int   s_idx[KNN];
  __shared__ float s_wpad[PADW ? COUT*C2P : 1];

  if (tid < KNN) s_idx[tid] = idx[(size_t)wg*KNN + tid];
  const float* cen = F + (size_t)wg*ldF + colF;
  for (int c = tid; c < CIN; c += NW*32) s_cen[c] = cen[c];
  __syncthreads();

  const float* baseF = F + (size_t)(b*NPTS)*ldF + colF;
  for (int t = tid; t < 32*C2P; t += NW*32) {
    int k = t / C2P, c = t - k*C2P;
    float v = 0.f;
    if (k < KNN && c < C2)
      v = (c < CIN) ? (baseF[(size_t)s_idx[k]*ldF + c] - s_cen[c]) : s_cen[c - CIN];
    s_edge[t] = v;
  }
  if (PADW) {
    for (int t = tid; t < COUT*C2P; t += NW*32) {
      int o = t / C2P, c = t - o*C2P;
      s_wpad[t] = (c < C2) ? W[(size_t)o*C2 + c] : 0.f;
    }
  }
  __syncthreads();

  int o0    = wv * 16;
  int mr    = lane & 15;
  int khalf = (lane >> 4) * 2;
  const float* wrow = W + (size_t)(o0 + mr)*C2;

  v8f acc0 = {};
  v8f acc1 = {};
  #pragma unroll 8
  for (int kk = 0; kk < C2P; kk += 4) {
    int c0 = kk + khalf;
    v2f a0 = *(const v2f*)&s_edge[(size_t)mr*C2P + c0];
    v2f a1 = *(const v2f*)&s_edge[(size_t)(16 + mr)*C2P + c0];
    v2f bb;
    if (PADW) bb = *(const v2f*)&s_wpad[(size_t)(o0 + mr)*C2P + c0];
    else      bb = *(const v2f*)(wrow + c0);
    acc0 = wmma_f32(a0, bb, acc0);
    acc1 = wmma_f32(a1, bb, acc1);
  }

  int o    = o0 + mr;
  float g  = bn[o];
  float be = bn[COUT + o];
  float mx = NEG_INF;
  int rbase = (lane >> 4) * 8;
  #pragma unroll
  for (int v = 0; v < 8; ++v) {
    float y0 = g * (acc0[v] * BN_INV_F) + be;
    mx = fmaxf(mx, LRELU(y0));
    float y1 = g * (acc1[v] * BN_INV_F) + be;
    y1 = LRELU(y1);
    int r1 = 16 + rbase + v;                   // branchless: select, not exec
    mx = fmaxf(mx, (r1 < KNN) ? y1 : NEG_INF);
  }
  mx = fmaxf(mx, __shfl_xor(mx, 16));
  if (lane < 16) Out[(size_t)wg*ldOut + colOut + o] = mx;
}

// ---------------------------------------------------------------------------
// Generic fp32-WMMA GEMM: C = epi(A(MxK) @ B) ; B is (K,N) or, if TRANSB,
// (N,K) row-major (i.e. A @ W^T). One wave per 16x16 tile, 4 waves/block.
// M rows are clamped on load and the C buffer is padded to a multiple of 16
// rows by the caller, so ALL loads and stores are unconditional.
// ---------------------------------------------------------------------------
enum { EPI_NONE = 0, EPI_BN_LRELU = 1, EPI_BIAS_RESID = 2, EPI_BIAS_GELU = 3 };

template<int EPI, bool TRANSB>
__global__ __launch_bounds__(128)
void gemm_kernel(const float* __restrict__ A, int lda,
                 const float* __restrict__ Bm, int ldb,
                 float* __restrict__ Cm, int ldc,
                 int M, int N, int Kd,
                 const float* __restrict__ p0, const float* __restrict__ p1)
{
  int lane = threadIdx.x & 31;
  int wv   = threadIdx.x >> 5;
  int m0   = (blockIdx.x * 4 + wv) * 16;
  int n0   = blockIdx.y * 16;
  if (m0 >= M) return;

  int mr = m0 + (lane & 15);
  if (mr >= M) mr = M - 1;                       // clamp: rows are independent
  const float* arow = A + (size_t)mr * lda;
  int khalf = (lane >> 4) * 2;
  int ncol  = n0 + (lane & 15);
  const float* brow = Bm + (TRANSB ? (size_t)ncol * ldb : 0);

  v8f acc = {};
  for (int k0 = 0; k0 < Kd; k0 += 64) {          // Kd is a multiple of 64
    __builtin_prefetch(arow + k0 + 64, 0, 1);
    #pragma unroll
    for (int k = k0; k < k0 + 64; k += 4) {
      int c0 = k + khalf;
      v2f a = *(const v2f*)(arow + c0);
      v2f bb;
      if (TRANSB) bb = *(const v2f*)(brow + c0);
      else { bb.x = Bm[(size_t)c0*ldb + ncol]; bb.y = Bm[(size_t)(c0+1)*ldb + ncol]; }
      acc = wmma_f32(a, bb, acc);
    }
  }

  int rbase = m0 + (lane >> 4) * 8;
  #pragma unroll
  for (int v = 0; v < 8; ++v) {
    int row = rbase + v;                         // C padded: no store guard
    float y = acc[v];
    if (EPI == EPI_BN_LRELU) {
      y = p0[ncol] * (y * BN_INV_F) + p1[ncol];
      y = LRELU(y);
    } else if (EPI == EPI_BIAS_GELU) {
      y += p0[ncol];
      y = 0.5f * y * (1.f + erff(y * 0.70710678118654752f));
    } else if (EPI == EPI_BIAS_RESID) {
      y += p0[ncol];
      y += Cm[(size_t)row*ldc + ncol];
    }
    Cm[(size_t)row*ldc + ncol] = y;
  }
}

// ---------------------------------------------------------------------------
// Patch max-pool: feats (B,2048,1024) -> h rows [b*65+1+p], max over 32 pts.
// ---------------------------------------------------------------------------
__global__ __launch_bounds__(256)
void patchmax_kernel(const float* __restrict__ feats, float* __restrict__ h)
{
  const int D = 1024;
  int bp = blockIdx.x;            // b*64 + p
  int b = bp >> 6, p = bp & 63;
  const float* src = feats + ((size_t)b*2048 + (size_t)p*32) * D;
  float* dst = h + ((size_t)(b*65) + 1 + p) * D;
  for (int c = threadIdx.x; c < D; c += 256) {
    float mx = src[c];
    #pragma unroll 4
    for (int j = 1; j < 32; ++j) mx = fmaxf(mx, src[(size_t)j*D + c]);
    dst[c] = mx;
  }
}

__global__ void init_cls_kernel(const float* __restrict__ cls, float* __restrict__ h)
{
  int i = blockIdx.x * blockDim.x + threadIdx.x;       // 2*1024
  if (i < 2048) {
    int b = i >> 10, c = i & 1023;
    h[(size_t)(b*65)*1024 + c] = cls[c];
  }
}

__global__ void add_pos_kernel(float* __restrict__ h, const float* __restrict__ pos, int n)
{
  int i = blockIdx.x * blockDim.x + threadIdx.x;
  if (i < n) h[i] += pos[i];
}

__global__ __launch_bounds__(256)
void layernorm_kernel(const float* __restrict__ X, const float* __restrict__ g,
                      const float* __restrict__ bt, float* __restrict__ Y)
{
  const int D = 1024;
  int row = blockIdx.x;
  const float* x = X + (size_t)row*D;
  float* y = Y + (size_t)row*D;
  int tid = threadIdx.x;
  float s = 0.f, s2 = 0.f;
  for (int c = tid; c < D; c += 256) { float v = x[c]; s += v; s2 += v*v; }
  __shared__ float sh1[256], sh2[256];
  sh1[tid] = s; sh2[tid] = s2;
  __syncthreads();
  for (int off = 128; off > 0; off >>= 1) {
    if (tid < off) { sh1[tid] += sh1[tid+off]; sh2[tid] += sh2[tid+off]; }
    __syncthreads();
  }
  float mean = sh1[0] * (1.f/D);
  float var  = sh2[0] * (1.f/D) - mean*mean;
  float inv  = rsqrtf(var + 1e-5f);
  for (int c = tid; c < D; c += 256) y[c] = (x[c] - mean) * inv * g[c] + bt[c];
}

// ---------------------------------------------------------------------------
// Attention core: per (b, head). L=65, d=64. K,V and scores in LDS.
// ---------------------------------------------------------------------------
__global__ __launch_bounds__(128)
void attention_kernel(const float* __restrict__ qkv, float* __restrict__ z)
{
  const int L = 65, DH = 64;
  int b = blockIdx.x >> 3, hh = blockIdx.x & 7;
  __shared__ float sk[L][DH];
  __shared__ float sv[L][DH];
  __shared__ float ss[L][L+1];
  int tid = threadIdx.x;
  const float* base = qkv + (size_t)(b*65)*1536 + hh*64;

  for (int t = tid; t < L*DH; t += 128) {
    int l = t >> 6, d = t & 63;
    sk[l][d] = base[(size_t)l*1536 + 512  + d];
    sv[l][d] = base[(size_t)l*1536 + 1024 + d];
  }
  __syncthreads();

  for (int t = tid; t < L*L; t += 128) {
    int l = t / L, m = t - l*L;
    const float* q = base + (size_t)l*1536;
    float s = 0.f;
    #pragma unroll 16
    for (int d = 0; d < DH; ++d) s += q[d] * sk[m][d];
    ss[l][m] = s * 0.125f;
  }
  __syncthreads();

  if (tid < L) {
    float mx = NEG_INF;
    for (int m = 0; m < L; ++m) mx = fmaxf(mx, ss[tid][m]);
    float sum = 0.f;
    for (int m = 0; m < L; ++m) { float e = __expf(ss[tid][m] - mx); ss[tid][m] = e; sum += e; }
    float inv = 1.f / sum;
    for (int m = 0; m < L; ++m) ss[tid][m] *= inv;
  }
  __syncthreads();

  for (int t = tid; t < L*DH; t += 128) {
    int l = t >> 6, d = t & 63;
    float a = 0.f;
    #pragma unroll 16
    for (int m = 0; m < L; ++m) a += ss[l][m] * sv[m][d];
    z[(size_t)(b*65 + l)*512 + hh*64 + d] = a;
  }
}

__global__ void extract_kernel(const float* __restrict__ h, float* __restrict__ out)
{
  int i = blockIdx.x * blockDim.x + threadIdx.x;       // 2*64*1024
  if (i < 131072) {
    int b = i >> 16;
    int rem = i & 65535;
    int l = rem >> 10, c = rem & 1023;
    out[i] = h[((size_t)(b*65) + 1 + l)*1024 + c];
  }
}

// ---------------------------------------------------------------------------
extern "C" void kernel_launch(void* const* d_in, const int* in_sizes, int n_in,
                              void* d_out, int out_size, void* d_ws, size_t ws_size,
                              hipStream_t stream)
{
  (void)in_sizes; (void)n_in; (void)out_size; (void)ws_size;
  const float* x    = (const float*)d_in[0];    // (2,2048,3)
  const float* pos  = (const float*)d_in[1];    // (2,65,1024)
  const float* cls  = (const float*)d_in[2];    // (1,1,1024)
  const float* w1   = (const float*)d_in[3];
  const float* bn1  = (const float*)d_in[4];
  const float* w2   = (const float*)d_in[5];
  const float* bn2  = (const float*)d_in[6];
  const float* w3   = (const float*)d_in[7];
  const float* bn3  = (const float*)d_in[8];
  const float* w4   = (const float*)d_in[9];
  const float* bn4  = (const float*)d_in[10];
  const float* w5   = (const float*)d_in[11];
  const float* bn5  = (const float*)d_in[12];
  const float* ln1  = (const float*)d_in[13];   // (6,2,1024)
  const float* wqkv = (const float*)d_in[14];   // (6,1024,1536)
  const float* wo   = (const float*)d_in[15];   // (6,512,1024)
  const float* bo   = (const float*)d_in[16];   // (6,1024)
  const float* ln2  = (const float*)d_in[17];   // (6,2,1024)
  const float* wf1  = (const float*)d_in[18];   // (6,1024,2048)
  const float* bf1  = (const float*)d_in[19];   // (6,2048)
  const float* wf2  = (const float*)d_in[20];   // (6,2048,1024)
  const float* bf2  = (const float*)d_in[21];   // (6,1024)

  // workspace carve-out (~29 MB); token buffers padded to 144 rows so GEMM
  // stores need no row guard (M=130 -> tiles cover rows 0..143).
  char* ws = (char*)d_ws;
  size_t off = 0;
  auto carve = [&](size_t bytes) -> char* {
    char* p = ws + off;
    off += (bytes + 255) & ~(size_t)255;
    return p;
  };
  const int MP = 144;                                    // padded token rows
  int*   idxb  = (int*)  carve((size_t)2*2048*20*sizeof(int));
  float* cat   = (float*)carve((size_t)2*2048*512*sizeof(float));   // x1|x2|x3|x4
  float* feats = (float*)carve((size_t)2*2048*1024*sizeof(float));
  float* h     = (float*)carve((size_t)MP*1024*sizeof(float));
  float* t     = (float*)carve((size_t)MP*1024*sizeof(float));
  float* qkvb  = (float*)carve((size_t)MP*1536*sizeof(float));
  float* zb    = (float*)carve((size_t)MP*512*sizeof(float));
  float* yb    = (float*)carve((size_t)MP*1024*sizeof(float));
  float* ub    = (float*)carve((size_t)MP*2048*sizeof(float));

  dim3 b32(32);
  // ---- DGCNN patch embed ----
  knn_kernel<3><<<4096, b32, 0, stream>>>(x, 3, 0, idxb);
  edgeconv_kernel<3,64><<<4096, 128, 0, stream>>>(x, 3, 0, idxb, w1, bn1, cat, 512, 0);
  knn_kernel<64><<<4096, b32, 0, stream>>>(cat, 512, 0, idxb);
  edgeconv_kernel<64,64><<<4096, 128, 0, stream>>>(cat, 512, 0, idxb, w2, bn2, cat, 512, 64);
  knn_kernel<64><<<4096, b32, 0, stream>>>(cat, 512, 64, idxb);
  edgeconv_kernel<64,128><<<4096, 256, 0, stream>>>(cat, 512, 64, idxb, w3, bn3, cat, 512, 128);
  knn_kernel<128><<<4096, b32, 0, stream>>>(cat, 512, 128, idxb);
  edgeconv_kernel<128,256><<<4096, 512, 0, stream>>>(cat, 512, 128, idxb, w4, bn4, cat, 512, 256);
  // conv5: feats = lrelu(bn5(cat @ w5^T)), (4096 x 1024)
  gemm_kernel<EPI_BN_LRELU, true><<<dim3(64, 64), 128, 0, stream>>>(
      cat, 512, w5, 512, feats, 1024, 4096, 1024, 512, bn5, bn5 + 1024);
  init_cls_kernel<<<8, 256, 0, stream>>>(cls, h);
  patchmax_kernel<<<128, 256, 0, stream>>>(feats, h);

  // ---- transformer ----
  const int M = 130;   // B*65 token rows (buffers padded to 144)
  for (int i = 0; i < 6; ++i) {
    add_pos_kernel<<<520, 256, 0, stream>>>(h, pos, 133120);
    layernorm_kernel<<<130, 256, 0, stream>>>(h, ln1 + (size_t)i*2048, ln1 + (size_t)i*2048 + 1024, t);
    gemm_kernel<EPI_NONE, false><<<dim3(3, 96), 128, 0, stream>>>(
        t, 1024, wqkv + (size_t)i*1024*1536, 1536, qkvb, 1536, M, 1536, 1024, nullptr, nullptr);
    attention_kernel<<<16, 128, 0, stream>>>(qkvb, zb);
    gemm_kernel<EPI_BIAS_RESID, false><<<dim3(3, 64), 128, 0, stream>>>(
        zb, 512, wo + (size_t)i*512*1024, 1024, h, 1024, M, 1024, 512, bo + (size_t)i*1024, nullptr);
    layernorm_kernel<<<130, 256, 0, stream>>>(h, ln2 + (size_t)i*2048, ln2 + (size_t)i*2048 + 1024, yb);
    gemm_kernel<EPI_BIAS_GELU, false><<<dim3(3, 128), 128, 0, stream>>>(
        yb, 1024, wf1 + (size_t)i*1024*2048, 2048, ub, 2048, M, 2048, 1024, bf1 + (size_t)i*2048, nullptr);
    gemm_kernel<EPI_BIAS_RESID, false><<<dim3(3, 64), 128, 0, stream>>>(
        ub, 2048, wf2 + (size_t)i*2048*1024, 1024, h, 1024, M, 1024, 2048, bf2 + (size_t)i*1024, nullptr);
  }
  extract_kernel<<<512, 256, 0, stream>>>(h, (float*)d_out);
}